// GCN_4483945857156
// MI455X (gfx1250) — compile-verified
//
#include <hip/hip_runtime.h>

// GCN forward: log_softmax( A@(relu(A@(xW1)+b1) W2) + b2 )
// N=50000 rows, E=800000 edges, dims 256 -> 128 -> 64, all fp32.
// GEMMs use V_WMMA_F32_16X16X4_F32 (native CDNA5 fp32 matrix op), one wave
// computes a 16x64 strip (4 independent accumulators -> A-fragment reuse + ILP).
// SpMM = edge-parallel gather + global_atomic_add_f32 scatter (L2-resident).

typedef __attribute__((ext_vector_type(2))) float v2f;
typedef __attribute__((ext_vector_type(8))) float v8f;

#define NROWS 50000
#define INDIM 256
#define HID   128
#define CLS   64
#define NEDGE 800000

// ---------------------------------------------------------------------------
// GEMM1: Y[50000,128] = X[50000,256] @ W1[256,128]
// block = 64 thr (2 waves, each a 16x64 strip), grid = 3125 M-tiles (exact).
// Per k-step: 1 A-fragment load feeds 4 independent WMMAs.
// ---------------------------------------------------------------------------
__global__ __launch_bounds__(64) void gcn_gemm1_wmma(
    const float* __restrict__ X, const float* __restrict__ W1,
    float* __restrict__ Y)
{
    const int mtile = blockIdx.x;            // 0..3124 (3125*16 = 50000 exact)
    const int wv    = threadIdx.x >> 5;      // 0..1 -> N half (64 cols each)
    const int lane  = threadIdx.x & 31;
    const int half  = lane >> 4;             // 0: K={0,1}, 1: K={2,3}
    const int l16   = lane & 15;

    const int m0 = mtile * 16;
    const int n0 = wv * 64;

    // A fragment source: row (m0+l16), cols k+2*half, k+2*half+1 (8B aligned)
    const float* xrow = X + (size_t)(m0 + l16) * INDIM + 2 * half;
    // B fragment base: col (n0 + 16*j + l16), rows k+2*half, k+2*half+1
    const float* wcol = W1 + n0 + l16;

    v8f acc0 = {}, acc1 = {}, acc2 = {}, acc3 = {};
    #pragma unroll 2
    for (int k = 0; k < INDIM; k += 4) {
        const float2 av = *(const float2*)(xrow + k);
        v2f a; a.x = av.x; a.y = av.y;
        const float* wk0 = wcol + (size_t)(k + 2 * half) * HID;
        const float* wk1 = wcol + (size_t)(k + 2 * half + 1) * HID;
        v2f b0; b0.x = wk0[0];  b0.y = wk1[0];
        v2f b1; b1.x = wk0[16]; b1.y = wk1[16];
        v2f b2; b2.x = wk0[32]; b2.y = wk1[32];
        v2f b3; b3.x = wk0[48]; b3.y = wk1[48];
        acc0 = __builtin_amdgcn_wmma_f32_16x16x4_f32(false, a, false, b0, (short)0, acc0, false, false);
        acc1 = __builtin_amdgcn_wmma_f32_16x16x4_f32(false, a, false, b1, (short)0, acc1, false, false);
        acc2 = __builtin_amdgcn_wmma_f32_16x16x4_f32(false, a, false, b2, (short)0, acc2, false, false);
        acc3 = __builtin_amdgcn_wmma_f32_16x16x4_f32(false, a, false, b3, (short)0, acc3, false, false);
    }

    // D layout: VGPR i -> row m0 + i + 8*half, col (n0 + 16*j + l16)
    float* yb = Y + n0 + l16;
    #pragma unroll
    for (int i = 0; i < 8; ++i) {
        float* yr = yb + (size_t)(m0 + i + 8 * half) * HID;
        yr[0]  = acc0[i];
        yr[16] = acc1[i];
        yr[32] = acc2[i];
        yr[48] = acc3[i];
    }
}

// ---------------------------------------------------------------------------
// GEMM2: Y[50000,64] = relu(Hraw[50000,128] + b1) @ W2[128,64]
// block = 32 thr (1 wave covers full N=64 with 4 accumulators), grid = 3125.
// ---------------------------------------------------------------------------
__global__ __launch_bounds__(32) void gcn_gemm2_wmma(
    const float* __restrict__ Hraw, const float* __restrict__ b1,
    const float* __restrict__ W2, float* __restrict__ Y)
{
    const int mtile = blockIdx.x;
    const int lane  = threadIdx.x & 31;
    const int half  = lane >> 4;
    const int l16   = lane & 15;

    const int m0 = mtile * 16;

    const float* hrow = Hraw + (size_t)(m0 + l16) * HID + 2 * half;
    const float* brow = b1 + 2 * half;
    const float* wcol = W2 + l16;

    v8f acc0 = {}, acc1 = {}, acc2 = {}, acc3 = {};
    #pragma unroll 2
    for (int k = 0; k < HID; k += 4) {
        const float2 hv = *(const float2*)(hrow + k);
        const float2 bv = *(const float2*)(brow + k);
        v2f a;
        a.x = fmaxf(hv.x + bv.x, 0.0f);      // relu(h + b1) fused into A load
        a.y = fmaxf(hv.y + bv.y, 0.0f);
        const float* wk0 = wcol + (size_t)(k + 2 * half) * CLS;
        const float* wk1 = wcol + (size_t)(k + 2 * half + 1) * CLS;
        v2f b0; b0.x = wk0[0];  b0.y = wk1[0];
        v2f b1v; b1v.x = wk0[16]; b1v.y = wk1[16];
        v2f b2v; b2v.x = wk0[32]; b2v.y = wk1[32];
        v2f b3v; b3v.x = wk0[48]; b3v.y = wk1[48];
        acc0 = __builtin_amdgcn_wmma_f32_16x16x4_f32(false, a, false, b0,  (short)0, acc0, false, false);
        acc1 = __builtin_amdgcn_wmma_f32_16x16x4_f32(false, a, false, b1v, (short)0, acc1, false, false);
        acc2 = __builtin_amdgcn_wmma_f32_16x16x4_f32(false, a, false, b2v, (short)0, acc2, false, false);
        acc3 = __builtin_amdgcn_wmma_f32_16x16x4_f32(false, a, false, b3v, (short)0, acc3, false, false);
    }

    float* yb = Y + l16;
    #pragma unroll
    for (int i = 0; i < 8; ++i) {
        float* yr = yb + (size_t)(m0 + i + 8 * half) * CLS;
        yr[0]  = acc0[i];
        yr[16] = acc1[i];
        yr[32] = acc2[i];
        yr[48] = acc3[i];
    }
}

// ---------------------------------------------------------------------------
// SpMM scatter, F=128: one wave per edge, lane -> 4 features (float4 gather).
// out must be pre-zeroed. Atomics resolve in L2 (dest 25.6 MB << 192 MB L2).
// ---------------------------------------------------------------------------
__global__ __launch_bounds__(256) void gcn_spmm_f128(
    const int* __restrict__ erow, const int* __restrict__ ecol,
    const float* __restrict__ eval, const float* __restrict__ dense,
    float* __restrict__ out)
{
    const int e = (int)((blockIdx.x * (size_t)blockDim.x + threadIdx.x) >> 5);
    const int lane = threadIdx.x & 31;
    if (e >= NEDGE) return;
    const int   r = erow[e];
    const int   c = ecol[e];
    const float v = eval[e];
    const float4 g = *(const float4*)(dense + (size_t)c * HID + lane * 4);
    float* o = out + (size_t)r * HID + lane * 4;
    atomicAdd(o + 0, v * g.x);
    atomicAdd(o + 1, v * g.y);
    atomicAdd(o + 2, v * g.z);
    atomicAdd(o + 3, v * g.w);
}

// SpMM scatter, F=64: one wave per edge, lane -> 2 features.
__global__ __launch_bounds__(256) void gcn_spmm_f64(
    const int* __restrict__ erow, const int* __restrict__ ecol,
    const float* __restrict__ eval, const float* __restrict__ dense,
    float* __restrict__ out)
{
    const int e = (int)((blockIdx.x * (size_t)blockDim.x + threadIdx.x) >> 5);
    const int lane = threadIdx.x & 31;
    if (e >= NEDGE) return;
    const int   r = erow[e];
    const int   c = ecol[e];
    const float v = eval[e];
    const float2 g = *(const float2*)(dense + (size_t)c * CLS + lane * 2);
    float* o = out + (size_t)r * CLS + lane * 2;
    atomicAdd(o + 0, v * g.x);
    atomicAdd(o + 1, v * g.y);
}

// ---------------------------------------------------------------------------
// log_softmax over rows of 64: one wave32 per row, 2 elements per lane.
// ---------------------------------------------------------------------------
__global__ __launch_bounds__(256) void gcn_logsoftmax64(
    const float* __restrict__ L, const float* __restrict__ b2,
    float* __restrict__ out)
{
    const int r = (int)((blockIdx.x * (size_t)blockDim.x + threadIdx.x) >> 5);
    const int lane = threadIdx.x & 31;
    if (r >= NROWS) return;

    float2 v = *(const float2*)(L + (size_t)r * CLS + lane * 2);
    const float2 bb = *(const float2*)(b2 + lane * 2);
    v.x += bb.x;
    v.y += bb.y;

    float m = fmaxf(v.x, v.y);
    #pragma unroll
    for (int off = 16; off > 0; off >>= 1)
        m = fmaxf(m, __shfl_xor(m, off, 32));

    float s = __expf(v.x - m) + __expf(v.y - m);
    #pragma unroll
    for (int off = 16; off > 0; off >>= 1)
        s += __shfl_xor(s, off, 32);

    const float lse = m + __logf(s);
    float2 o;
    o.x = v.x - lse;
    o.y = v.y - lse;
    *(float2*)(out + (size_t)r * CLS + lane * 2) = o;
}

__global__ __launch_bounds__(256) void gcn_zero_f32(float* __restrict__ p, int n)
{
    const int i = (int)(blockIdx.x * (size_t)blockDim.x + threadIdx.x);
    if (i < n) p[i] = 0.0f;
}

// ---------------------------------------------------------------------------
// Host-side orchestration (all launches on `stream`, graph-capture safe).
// Workspace layout (peak live 51.2 MB):
//   slot0 = ws + 0        : XW1 [50000,128]  -> later reused as HW2 [50000,64]
//   slot1 = ws + 6400000  : h_raw[50000,128] -> later reused as logits[50000,64]
// ---------------------------------------------------------------------------
extern "C" void kernel_launch(void* const* d_in, const int* in_sizes, int n_in,
                              void* d_out, int out_size, void* d_ws, size_t ws_size,
                              hipStream_t stream) {
    const float* x        = (const float*)d_in[0];
    const int*   edge_row = (const int*)  d_in[1];
    const int*   edge_col = (const int*)  d_in[2];
    const float* edge_val = (const float*)d_in[3];
    const float* W1       = (const float*)d_in[4];
    const float* b1       = (const float*)d_in[5];
    const float* W2       = (const float*)d_in[6];
    const float* b2       = (const float*)d_in[7];
    float* out = (float*)d_out;

    float* ws    = (float*)d_ws;
    float* XW1   = ws;                       // 50000*128 floats = 25.6 MB
    float* Hraw  = ws + (size_t)NROWS * HID; // 25.6 MB
    float* HW2   = XW1;                      // 50000*64 (XW1 dead after spmm1)
    float* Lraw  = Hraw;                     // 50000*64 (Hraw dead after gemm2)

    const int MT = NROWS / 16;               // 3125 exact M-tiles
    const int edge_blocks = (NEDGE * 32 + 255) / 256;   // 1 wave per edge

    // 1) XW1 = x @ W1               (WMMA fp32, 16x64 strip per wave)
    gcn_gemm1_wmma<<<MT, 64, 0, stream>>>(x, W1, XW1);
    // 2) h_raw = A @ XW1            (zero + atomic scatter)
    gcn_zero_f32<<<(NROWS * HID + 255) / 256, 256, 0, stream>>>(Hraw, NROWS * HID);
    gcn_spmm_f128<<<edge_blocks, 256, 0, stream>>>(edge_row, edge_col, edge_val, XW1, Hraw);
    // 3) HW2 = relu(h_raw + b1) @ W2  (WMMA fp32, relu+bias fused into A load)
    gcn_gemm2_wmma<<<MT, 32, 0, stream>>>(Hraw, b1, W2, HW2);
    // 4) logits = A @ HW2
    gcn_zero_f32<<<(NROWS * CLS + 255) / 256, 256, 0, stream>>>(Lraw, NROWS * CLS);
    gcn_spmm_f64<<<edge_blocks, 256, 0, stream>>>(edge_row, edge_col, edge_val, HW2, Lraw);
    // 5) out = log_softmax(logits + b2)
    gcn_logsoftmax64<<<(NROWS * 32 + 255) / 256, 256, 0, stream>>>(Lraw, b2, out);
}